// LightGCN_22041772163102
// MI455X (gfx1250) — compile-verified
//
#include <hip/hip_runtime.h>
#include <hip/hip_bf16.h>

#define NU 100000
#define NI 50000
#define NN 150000
#define ED 64
#define NE 1500000
#define NB 4096

typedef __attribute__((ext_vector_type(2))) float v2f;
typedef __attribute__((ext_vector_type(8))) float v8f;

// ---------------- workspace layout (float offsets) ----------------
// deg/dinv : [0,        150000)
// norm     : [150000,   1650000)
// repA     : [1650000,  11250000)
// repB     : [11250000, 20850000)
// acc      : [20850000, 30450000)   -> ~121.8 MB total

__device__ __forceinline__ void atomAddF(float* p, float v) {
  unsafeAtomicAdd(p, v);   // hardware global_atomic_add_f32
}

__global__ void lgcn_zero_deg(unsigned* __restrict__ deg) {
  int i = blockIdx.x * blockDim.x + threadIdx.x;
  if (i < NN) deg[i] = 0u;
}

__global__ void lgcn_count(const int* __restrict__ eu, const int* __restrict__ ei,
                           unsigned* __restrict__ deg) {
  int k = blockIdx.x * blockDim.x + threadIdx.x;
  if (k < NE) {
    atomicAdd(&deg[eu[k]], 1u);
    atomicAdd(&deg[NU + ei[k]], 1u);
  }
}

__global__ void lgcn_dinv(float* __restrict__ dinv) {
  int i = blockIdx.x * blockDim.x + threadIdx.x;
  if (i < NN) {
    unsigned c = ((const unsigned*)dinv)[i];
    float d = (c > 0u) ? (float)c : 1.0f;
    dinv[i] = 1.0f / sqrtf(d);
  }
}

__global__ void lgcn_norm(const int* __restrict__ eu, const int* __restrict__ ei,
                          const float* __restrict__ dinv, float* __restrict__ nrm) {
  int k = blockIdx.x * blockDim.x + threadIdx.x;
  if (k < NE) nrm[k] = dinv[eu[k]] * dinv[NU + ei[k]];
}

__global__ void lgcn_init(const float4* __restrict__ emb, float4* __restrict__ repA,
                          float4* __restrict__ repB, float4* __restrict__ acc) {
  int i = blockIdx.x * blockDim.x + threadIdx.x;
  if (i < NN * (ED / 4)) {
    float4 e = emb[i];
    repA[i] = e;
    acc[i] = e;
    repB[i] = make_float4(0.f, 0.f, 0.f, 0.f);
  }
}

// One wave per interaction. Lanes 0-15: rep[u] -> next[item]; lanes 16-31: rep[item] -> next[u].
// Each lane: one b128 load + 4 fp32 global atomic adds (covers all 64 dims, both directions).
__global__ void lgcn_scatter(const int* __restrict__ eu, const int* __restrict__ ei,
                             const float* __restrict__ nrm,
                             const float* __restrict__ rep, float* __restrict__ nxt) {
  int gid = blockIdx.x * blockDim.x + threadIdx.x;
  int k = gid >> 5;
  if (k >= NE) return;
  int lane = threadIdx.x & 31;
  int half = lane >> 4;
  int l16 = lane & 15;
  int u = eu[k];
  int it = NU + ei[k];
  float w = nrm[k];
  int src = half ? it : u;
  int dst = half ? u : it;
  const float4* s = (const float4*)(rep + (size_t)src * ED);
  float4 v = s[l16];
  float* dp = nxt + (size_t)dst * ED + (l16 << 2);
  atomAddF(dp + 0, v.x * w);
  atomAddF(dp + 1, v.y * w);
  atomAddF(dp + 2, v.z * w);
  atomAddF(dp + 3, v.w * w);
}

// acc += rep_next; optionally zero the just-consumed rep buffer for the next layer.
__global__ void lgcn_update(const float4* __restrict__ nxt, float4* __restrict__ acc,
                            float4* __restrict__ oldrep, int zero_old) {
  int i = blockIdx.x * blockDim.x + threadIdx.x;
  if (i < NN * (ED / 4)) {
    float4 v = nxt[i];
    float4 a = acc[i];
    a.x += v.x; a.y += v.y; a.z += v.z; a.w += v.w;
    acc[i] = a;
    if (zero_old) oldrep[i] = make_float4(0.f, 0.f, 0.f, 0.f);
  }
}

// users_r / pos_items_r / neg_items_r = acc[row]/4, vectorized float4.
__global__ void lgcn_gather(const float4* __restrict__ acc, const int* __restrict__ users,
                            const int* __restrict__ pos, const int* __restrict__ neg,
                            float4* __restrict__ out) {
  int i = blockIdx.x * blockDim.x + threadIdx.x;
  if (i >= 3 * NB * (ED / 4)) return;
  int which = i / (NB * (ED / 4));
  int r = i - which * (NB * (ED / 4));
  int b = r >> 4;
  int c = r & 15;
  int row = (which == 0) ? users[b] : ((which == 1) ? (NU + pos[b]) : (NU + neg[b]));
  float4 v = acc[(size_t)row * (ED / 4) + c];
  v.x *= 0.25f; v.y *= 0.25f; v.z *= 0.25f; v.w *= 0.25f;
  out[i] = v;
}

// l2_norm_sq via fp32 WMMA Gram trick: for V_WMMA_F32_16X16X4_F32 the A(16x4) and
// B=A^T(4x16) VGPR layouts coincide, so D = X X^T chunk; diagonal = row sum-of-squares.
// One wave handles 16 batch rows; 3 tables x 16 K-chunks = 48 chained WMMAs.
__global__ void lgcn_l2_wmma(const float* __restrict__ emb, const int* __restrict__ users,
                             const int* __restrict__ pos, const int* __restrict__ neg,
                             float* __restrict__ l2out) {
  int tid = blockIdx.x * blockDim.x + threadIdx.x;
  int wave = tid >> 5;           // 0..255, 16 batch rows each
  int lane = threadIdx.x & 31;
  int half = lane >> 4;          // K sub-chunk select (cols 4j+0,1 vs 4j+2,3)
  int l16 = lane & 15;           // matrix row M
  int b = (wave << 4) + l16;

  int rows0 = users[b];
  int rows1 = NU + pos[b];
  int rows2 = NU + neg[b];

  v8f c = {0.f, 0.f, 0.f, 0.f, 0.f, 0.f, 0.f, 0.f};
  int rows[3] = {rows0, rows1, rows2};
#pragma unroll
  for (int t = 0; t < 3; ++t) {
    const v2f* p = (const v2f*)(emb + (size_t)rows[t] * ED);
#pragma unroll
    for (int j = 0; j < 16; ++j) {
      v2f a = p[2 * j + half];   // lane holds X[M=l16, K=4j+2*half .. +1]
      c = __builtin_amdgcn_wmma_f32_16x16x4_f32(
          /*neg_a=*/false, a, /*neg_b=*/false, a,
          /*c_mod=*/(short)0, c, /*reuse_a=*/false, /*reuse_b=*/false);
    }
  }

  // Diagonal extraction: lanes 0-7 hold M=N=lane at vgpr[lane];
  // lanes 24-31 hold M=N=lane-16 at vgpr[lane-24] (== lane&7).
  int s = lane & 7;
  float d = c[0];
  d = (s == 1) ? c[1] : d;
  d = (s == 2) ? c[2] : d;
  d = (s == 3) ? c[3] : d;
  d = (s == 4) ? c[4] : d;
  d = (s == 5) ? c[5] : d;
  d = (s == 6) ? c[6] : d;
  d = (s == 7) ? c[7] : d;
  if (lane < 8)        l2out[(wave << 4) + lane] = d;
  else if (lane >= 24) l2out[(wave << 4) + lane - 16] = d;
}

extern "C" void kernel_launch(void* const* d_in, const int* in_sizes, int n_in,
                              void* d_out, int out_size, void* d_ws, size_t ws_size,
                              hipStream_t stream) {
  const float* emb = (const float*)d_in[0];
  const int* eu = (const int*)d_in[1];
  const int* ei = (const int*)d_in[2];
  const int* users = (const int*)d_in[3];
  const int* pos = (const int*)d_in[4];
  const int* neg = (const int*)d_in[5];
  float* out = (float*)d_out;

  float* ws = (float*)d_ws;
  float* deg = ws;                 // uint during count, float dinv after
  float* nrm = ws + 150000;
  float* repA = ws + 1650000;
  float* repB = ws + 11250000;
  float* acc = ws + 20850000;

  const int B = 256;
  lgcn_zero_deg<<<(NN + B - 1) / B, B, 0, stream>>>((unsigned*)deg);
  lgcn_count<<<(NE + B - 1) / B, B, 0, stream>>>(eu, ei, (unsigned*)deg);
  lgcn_dinv<<<(NN + B - 1) / B, B, 0, stream>>>(deg);
  lgcn_norm<<<(NE + B - 1) / B, B, 0, stream>>>(eu, ei, deg, nrm);

  const int NV4 = NN * (ED / 4);   // 2.4M float4 per node matrix
  lgcn_init<<<(NV4 + B - 1) / B, B, 0, stream>>>((const float4*)emb, (float4*)repA,
                                                 (float4*)repB, (float4*)acc);

  float* cur = repA;
  float* nxt = repB;
  for (int l = 0; l < 3; ++l) {
    // NE waves, 32 threads each: 48M threads / 256 = 187500 blocks (exact)
    lgcn_scatter<<<(NE * 32) / B, B, 0, stream>>>(eu, ei, nrm, cur, nxt);
    lgcn_update<<<(NV4 + B - 1) / B, B, 0, stream>>>((const float4*)nxt, (float4*)acc,
                                                     (float4*)cur, (l < 2) ? 1 : 0);
    float* t = cur; cur = nxt; nxt = t;
  }

  lgcn_gather<<<(3 * NB * (ED / 4) + B - 1) / B, B, 0, stream>>>(
      (const float4*)acc, users, pos, neg, (float4*)out);

  // 256 waves = 32 blocks x 256 threads; writes l2_norm_sq after the 3 rep outputs
  lgcn_l2_wmma<<<32, B, 0, stream>>>(emb, users, pos, neg, out + 3 * NB * ED);
}